// SparseAttentionMaskGenerator_82892868813320
// MI455X (gfx1250) — compile-verified
//
#include <hip/hip_runtime.h>
#include <stdint.h>

// Problem constants (B=1, H=16, S=2048)
#define HEADS 16
#define SEQ   2048
#define NPH   (SEQ * SEQ)          // 4,194,304 elements per head
#define BINS  65536                // 16-bit radix histogram
#define BPH   16                   // streaming blocks per head
#define TPB   256                  // 8 wave32 per block
#define CHUNK (NPH / BPH)          // 262,144 floats per block
#define ITERS (CHUNK / (TPB * 4))  // 256 iterations, 16B per lane per iter
#define NBUF  8                    // async pipeline depth (ring of LDS buffers)
// jnp.quantile(0.95): pos = 0.95*(N-1) = 3984587.85
#define K0 3984587u
#define K1 3984588u
#define FRAC 0.85f

typedef __attribute__((ext_vector_type(4))) float v4f;

__device__ __forceinline__ uint32_t bits2key(uint32_t u) {
    // monotone float -> uint mapping (handles negatives)
    return (u & 0x80000000u) ? ~u : (u | 0x80000000u);
}
__device__ __forceinline__ float key2f(uint32_t k) {
    uint32_t u = (k & 0x80000000u) ? (k & 0x7FFFFFFFu) : ~k;
    return __uint_as_float(u);
}

// wait until at most `n` async ops outstanding (immediate must be literal)
__device__ __forceinline__ void wait_async_upto(int n) {
    switch (n) {
        case 0: asm volatile("s_wait_asynccnt 0" ::: "memory"); break;
        case 1: asm volatile("s_wait_asynccnt 1" ::: "memory"); break;
        case 2: asm volatile("s_wait_asynccnt 2" ::: "memory"); break;
        case 3: asm volatile("s_wait_asynccnt 3" ::: "memory"); break;
        case 4: asm volatile("s_wait_asynccnt 4" ::: "memory"); break;
        case 5: asm volatile("s_wait_asynccnt 5" ::: "memory"); break;
        case 6: asm volatile("s_wait_asynccnt 6" ::: "memory"); break;
        default: asm volatile("s_wait_asynccnt 7" ::: "memory"); break;
    }
}

__device__ __forceinline__ void async_load_16B(uint32_t lds_addr, uint32_t voff,
                                               unsigned long long sbase) {
    // GVS mode: mem_addr = SGPR64 + VGPR32 offset; LDS dst addr per lane
    asm volatile("global_load_async_to_lds_b128 %0, %1, %2"
                 :: "v"(lds_addr), "v"(voff), "s"(sbase) : "memory");
}

// -------- Pass 1 / Pass 3: streaming radix histogram via async global->LDS ----
// MODE 0: histogram key>>16.  MODE 1: histogram key&0xFFFF restricted to hi-bin
// b0 (from binfo), plus atomicMin of the smallest key strictly above bin b0.
template <int MODE>
__global__ __launch_bounds__(TPB) void stream_hist(
    const float* __restrict__ scores, uint32_t* __restrict__ ghist,
    const uint32_t* __restrict__ binfo, uint32_t* __restrict__ min_above) {
    // CDNA5: 320KB LDS per WG -> 256KB histogram + 32KB async staging ring
    __shared__ uint32_t hist[BINS];          // 256 KB
    __shared__ uint4 stage[NBUF][TPB];       // 32 KB, 8-deep per-wave pipeline

    const int tid  = threadIdx.x;
    const int head = blockIdx.x / BPH;
    const int sub  = blockIdx.x % BPH;

    for (int b = tid; b < BINS; b += TPB) hist[b] = 0;
    __syncthreads();

    const unsigned long long hbase =
        (unsigned long long)(uintptr_t)(scores + (size_t)head * NPH);
    const uint32_t base_voff =
        (uint32_t)(((size_t)sub * CHUNK + (size_t)tid * 4) * sizeof(float));
    uint32_t ldsb[NBUF];
#pragma unroll
    for (int p = 0; p < NBUF; ++p)
        ldsb[p] = (uint32_t)(uintptr_t)&stage[p][tid];

    uint32_t b0 = 0;
    if (MODE == 1) b0 = binfo[head * 4 + 0];
    uint32_t lmin = 0xFFFFFFFFu;

    // prime NBUF-1 buffers: each lane async-copies 16B global -> its LDS slot
#pragma unroll
    for (int p = 0; p < NBUF - 1; ++p)
        async_load_16B(ldsb[p],
                       base_voff + (uint32_t)(p * TPB * 4 * sizeof(float)),
                       hbase);

    for (int it = 0; it < ITERS; ++it) {
        const int cur = it & (NBUF - 1);
        int ahead = ITERS - 1 - it;          // buffers issued beyond 'it'
        if (it + NBUF - 1 < ITERS) {
            const uint32_t voff = base_voff +
                (uint32_t)((it + NBUF - 1) * TPB * 4 * sizeof(float));
            async_load_16B(ldsb[(it + NBUF - 1) & (NBUF - 1)], voff, hbase);
            ahead = NBUF - 1;
        }
        // async loads of one wave complete in order: <=ahead outstanding
        // means buffer 'cur' has landed in LDS
        wait_async_upto(ahead);

        const uint4 d = stage[cur][tid];
        const uint32_t w[4] = {d.x, d.y, d.z, d.w};
#pragma unroll
        for (int j = 0; j < 4; ++j) {
            const uint32_t key = bits2key(w[j]);
            if (MODE == 0) {
                atomicAdd(&hist[key >> 16], 1u);   // ds_add_u32
            } else {
                const uint32_t hi = key >> 16;
                if (hi == b0)
                    atomicAdd(&hist[key & 0xFFFFu], 1u);
                else if (hi > b0 && key < lmin)
                    lmin = key;
            }
        }
    }

    if (MODE == 1 && lmin != 0xFFFFFFFFu) atomicMin(&min_above[head], lmin);
    __syncthreads();
    for (int b = tid; b < BINS; b += TPB) {
        const uint32_t c = hist[b];
        if (c) atomicAdd(&ghist[(size_t)head * BINS + b], c);
    }
}

// -------- Pass 2: per-head scan of hi histogram -> bins+residual ranks --------
__global__ __launch_bounds__(TPB) void scan_hi_kernel(
    const uint32_t* __restrict__ ghist, uint32_t* __restrict__ binfo) {
    __shared__ uint32_t ps[TPB];
    const int tid = threadIdx.x;
    const int head = blockIdx.x;
    const uint32_t* h = ghist + (size_t)head * BINS;
    const int SEG = BINS / TPB;  // 256 bins per thread

    uint32_t s = 0;
    for (int i = 0; i < SEG; ++i) s += h[tid * SEG + i];
    ps[tid] = s;
    __syncthreads();
    for (int off = 1; off < TPB; off <<= 1) {  // Hillis-Steele inclusive scan
        const uint32_t t = (tid >= off) ? ps[tid - off] : 0;
        __syncthreads();
        ps[tid] += t;
        __syncthreads();
    }
    uint32_t cum = (tid == 0) ? 0u : ps[tid - 1];
    for (int i = 0; i < SEG; ++i) {
        const uint32_t c = h[tid * SEG + i];
        const uint32_t lo = cum, hi = cum + c;
        if (K0 >= lo && K0 < hi) {
            binfo[head * 4 + 0] = (uint32_t)(tid * SEG + i);
            binfo[head * 4 + 1] = K0 - lo;
        }
        if (K1 >= lo && K1 < hi) {
            binfo[head * 4 + 2] = (uint32_t)(tid * SEG + i);
            binfo[head * 4 + 3] = K1 - lo;
        }
        cum = hi;
    }
}

// -------- Pass 4: per-head scan of lo histogram -> exact threshold ------------
__global__ __launch_bounds__(TPB) void scan_lo_kernel(
    const uint32_t* __restrict__ ghist_lo, const uint32_t* __restrict__ binfo,
    const uint32_t* __restrict__ min_above, float* __restrict__ thr) {
    __shared__ uint32_t ps[TPB];
    __shared__ float sv0, sv1;
    const int tid = threadIdx.x;
    const int head = blockIdx.x;
    const uint32_t b0 = binfo[head * 4 + 0], r0 = binfo[head * 4 + 1];
    const uint32_t b1 = binfo[head * 4 + 2], r1 = binfo[head * 4 + 3];
    if (tid == 0 && b1 != b0) sv1 = key2f(min_above[head]);  // next order stat
                                                             // lies above bin b0
    const uint32_t* h = ghist_lo + (size_t)head * BINS;
    const int SEG = BINS / TPB;
    uint32_t s = 0;
    for (int i = 0; i < SEG; ++i) s += h[tid * SEG + i];
    ps[tid] = s;
    __syncthreads();
    for (int off = 1; off < TPB; off <<= 1) {
        const uint32_t t = (tid >= off) ? ps[tid - off] : 0;
        __syncthreads();
        ps[tid] += t;
        __syncthreads();
    }
    uint32_t cum = (tid == 0) ? 0u : ps[tid - 1];
    for (int i = 0; i < SEG; ++i) {
        const uint32_t c = h[tid * SEG + i];
        const uint32_t lo = cum, hi = cum + c;
        if (r0 >= lo && r0 < hi)
            sv0 = key2f((b0 << 16) | (uint32_t)(tid * SEG + i));
        if (b1 == b0 && r1 >= lo && r1 < hi)
            sv1 = key2f((b0 << 16) | (uint32_t)(tid * SEG + i));
        cum = hi;
    }
    __syncthreads();
    if (tid == 0) thr[head] = sv0 + FRAC * (sv1 - sv0);  // linear interpolation
}

// -------- Pass 5: mask = scores >= thr[head], v4f + NT stores -----------------
__global__ __launch_bounds__(TPB) void apply_mask(
    const v4f* __restrict__ scores, v4f* __restrict__ out,
    const float* __restrict__ thr) {
    const size_t total4 = (size_t)HEADS * NPH / 4;
    for (size_t i = (size_t)blockIdx.x * blockDim.x + threadIdx.x; i < total4;
         i += (size_t)gridDim.x * blockDim.x) {
        const int head = (int)((i * 4) >> 22);  // NPH == 2^22
        const float t = thr[head];
        const v4f sv = scores[i];
        v4f o;
        o.x = (sv.x >= t) ? 1.0f : 0.0f;
        o.y = (sv.y >= t) ? 1.0f : 0.0f;
        o.z = (sv.z >= t) ? 1.0f : 0.0f;
        o.w = (sv.w >= t) ? 1.0f : 0.0f;
        // write-once output: non-temporal so it doesn't evict scores from L2
        __builtin_nontemporal_store(o, &out[i]);
    }
}

extern "C" void kernel_launch(void* const* d_in, const int* in_sizes, int n_in,
                              void* d_out, int out_size, void* d_ws,
                              size_t ws_size, hipStream_t stream) {
    const float* scores = (const float*)d_in[0];

    // workspace layout (~8.4 MB): hist_hi | hist_lo | binfo | min_above | thr
    uint32_t* hist_hi = (uint32_t*)d_ws;
    uint32_t* hist_lo = hist_hi + (size_t)HEADS * BINS;
    uint32_t* binfo   = hist_lo + (size_t)HEADS * BINS;
    uint32_t* min_ab  = binfo + HEADS * 4;
    float*    thr     = (float*)(min_ab + HEADS);

    (void)hipMemsetAsync(hist_hi, 0,
                         (size_t)2 * HEADS * BINS * sizeof(uint32_t), stream);
    (void)hipMemsetAsync(min_ab, 0xFF, HEADS * sizeof(uint32_t), stream);

    const dim3 sgrid(HEADS * BPH), sblk(TPB);
    stream_hist<0><<<sgrid, sblk, 0, stream>>>(scores, hist_hi, binfo, min_ab);
    scan_hi_kernel<<<HEADS, TPB, 0, stream>>>(hist_hi, binfo);
    stream_hist<1><<<sgrid, sblk, 0, stream>>>(scores, hist_lo, binfo, min_ab);
    scan_lo_kernel<<<HEADS, TPB, 0, stream>>>(hist_lo, binfo, min_ab, thr);
    apply_mask<<<4096, TPB, 0, stream>>>((const v4f*)scores, (v4f*)d_out, thr);
}